// GCN_61856118997368
// MI455X (gfx1250) — compile-verified
//
#include <hip/hip_runtime.h>

// N = 100000 nodes, E = 1600000 edges, F_IN = 128, H = 64, C = 16
// d_out = [out (N*16) | x_latent (N*64)] as float32.

typedef __attribute__((ext_vector_type(2))) float v2f;
typedef __attribute__((ext_vector_type(8))) float v8f;

__device__ __forceinline__ void atom_add_f32(float* p, float v) {
    __hip_atomic_fetch_add(p, v, __ATOMIC_RELAXED, __HIP_MEMORY_SCOPE_AGENT);
}

// ---------------------------------------------------------------- utilities
__global__ __launch_bounds__(256) void zero_f32(float* __restrict__ p, int n) {
    int i = blockIdx.x * blockDim.x + threadIdx.x;
    if (i < n) p[i] = 0.0f;
}

// deg[dst] += 1 per edge
__global__ __launch_bounds__(256) void degree_kernel(const int* __restrict__ dst,
                                                     float* __restrict__ deg, int E) {
    int i = blockIdx.x * blockDim.x + threadIdx.x;
    if (i < E) atom_add_f32(&deg[dst[i]], 1.0f);
}

// in-place: deg -> rsqrt(deg + 1)   (implicit self loop)
__global__ __launch_bounds__(256) void dinv_kernel(float* __restrict__ deg, int n) {
    int i = blockIdx.x * blockDim.x + threadIdx.x;
    if (i < n) deg[i] = rsqrtf(deg[i] + 1.0f);
}

// ---------------------------------------------------------------- WMMA GEMM
// H[m0:m0+16, n0:n0+16] = X[m0:m0+16, :] @ W[:, n0:n0+16]
// One wave per 16x16 tile, K-steps of 4 via V_WMMA_F32_16X16X4_F32.
template <int KDIM, int FOUT>
__global__ __launch_bounds__(32) void gemm_wmma_f32(const float* __restrict__ X,
                                                    const float* __restrict__ W,
                                                    float* __restrict__ Hout) {
    __shared__ float As[16 * (KDIM + 1)];   // +1 pad: stride-KDIM column reads
                                            // would all land in one LDS bank
    const int lane = threadIdx.x;           // 0..31 (wave32)
    const int m0 = blockIdx.x * 16;
    const int n0 = blockIdx.y * 16;

    // Cooperative, coalesced load of the 16 x KDIM A tile.
    for (int idx = lane; idx < 16 * KDIM; idx += 32) {
        int r = idx / KDIM;                 // KDIM is a power of two -> shifts
        int c = idx - r * KDIM;
        As[r * (KDIM + 1) + c] = X[(m0 + r) * KDIM + c];
    }
    __syncthreads();

    const int half = lane >> 4;             // 0: K=k,k+1  1: K=k+2,k+3
    const int l16  = lane & 15;

    v8f acc = {};
#pragma unroll
    for (int k = 0; k < KDIM; k += 4) {
        const int ka = k + 2 * half;
        v2f a, b;
        // A fragment: 32-bit A 16x4 layout (row = l16, K split across halves)
        a.x = As[l16 * (KDIM + 1) + ka];
        a.y = As[l16 * (KDIM + 1) + ka + 1];
        // B fragment: rows of W striped across lanes (N = n0 + l16)
        b.x = W[ka * FOUT + n0 + l16];
        b.y = W[(ka + 1) * FOUT + n0 + l16];
        acc = __builtin_amdgcn_wmma_f32_16x16x4_f32(
            /*neg_a=*/false, a, /*neg_b=*/false, b,
            /*c_mod=*/(short)0, acc, /*reuse_a=*/false, /*reuse_b=*/false);
    }

    // C/D layout: VGPR v on lane holds D[v + 8*half][l16]
#pragma unroll
    for (int v = 0; v < 8; ++v) {
        Hout[(m0 + v + 8 * half) * FOUT + n0 + l16] = acc[v];
    }
}

// ------------------------------------------------------------ edge scatter
// agg[dst] += h[src] * (dinv[src]*dinv[dst]) ; one wave per edge (F = 64)
__global__ __launch_bounds__(256) void scatter64(const float* __restrict__ h,
                                                 const int* __restrict__ src,
                                                 const int* __restrict__ dst,
                                                 const float* __restrict__ dinv,
                                                 float* __restrict__ agg, int E) {
    int wid  = (blockIdx.x * blockDim.x + threadIdx.x) >> 5;
    int lane = threadIdx.x & 31;
    if (wid >= E) return;
    int s = src[wid], d = dst[wid];
    float norm = dinv[s] * dinv[d];
    const float* hs = h + (size_t)s * 64;
    float* ad = agg + (size_t)d * 64;
    atom_add_f32(&ad[lane],      hs[lane]      * norm);
    atom_add_f32(&ad[lane + 32], hs[lane + 32] * norm);
}

// F = 16: 16 threads per edge
__global__ __launch_bounds__(256) void scatter16(const float* __restrict__ h,
                                                 const int* __restrict__ src,
                                                 const int* __restrict__ dst,
                                                 const float* __restrict__ dinv,
                                                 float* __restrict__ agg, int E) {
    int gid = blockIdx.x * blockDim.x + threadIdx.x;
    int e = gid >> 4;
    if (e >= E) return;
    int f = gid & 15;
    int s = src[e], d = dst[e];
    float norm = dinv[s] * dinv[d];
    atom_add_f32(&agg[(size_t)d * 16 + f], h[(size_t)s * 16 + f] * norm);
}

// ------------------------------------------------------------- finalize
// out = [relu]( agg + h * dinv^2 + bias )
template <int F, bool RELU>
__global__ __launch_bounds__(256) void finalize_kernel(const float* __restrict__ agg,
                                                       const float* __restrict__ h,
                                                       const float* __restrict__ dinv,
                                                       const float* __restrict__ bias,
                                                       float* __restrict__ out, int n) {
    int i = blockIdx.x * blockDim.x + threadIdx.x;
    if (i >= n * F) return;
    int node = i >> (F == 64 ? 6 : 4);
    int f    = i & (F - 1);
    float di = dinv[node];
    float v = agg[i] + h[i] * (di * di) + bias[f];
    if (RELU) v = fmaxf(v, 0.0f);
    out[i] = v;
}

// ------------------------------------------------------------- launch
extern "C" void kernel_launch(void* const* d_in, const int* in_sizes, int n_in,
                              void* d_out, int out_size, void* d_ws, size_t ws_size,
                              hipStream_t stream) {
    const float* x  = (const float*)d_in[0];   // [N,128]
    const int*   ei = (const int*)  d_in[1];   // [2,E]
    const float* W1 = (const float*)d_in[2];   // [128,64]
    const float* b1 = (const float*)d_in[3];   // [64]
    const float* Wh = (const float*)d_in[4];   // [64,64]
    const float* bh = (const float*)d_in[5];   // [64]
    const float* W2 = (const float*)d_in[6];   // [64,16]
    const float* b2 = (const float*)d_in[7];   // [16]

    const int N = in_sizes[0] / 128;           // 100000 (divisible by 16)
    const int E = in_sizes[1] / 2;             // 1600000
    const int* src = ei;
    const int* dst = ei + E;

    float* out  = (float*)d_out;               // [N,16]
    float* xlat = out + (size_t)N * 16;        // [N,64] (x_latent = h2)

    float* ws   = (float*)d_ws;
    float* dinv = ws;                          // [N]
    float* hbuf = dinv + N;                    // [N,64]  (gemm output, reused)
    float* agg  = hbuf + (size_t)N * 64;       // [N,64]
    float* l1   = agg  + (size_t)N * 64;       // [N,64]

    const int B = 256;
    auto blocks = [](long long n, int b) { return (unsigned)((n + b - 1) / b); };

    // ---- degree / normalization (recomputed every call: deterministic) ----
    zero_f32<<<blocks(N, B), B, 0, stream>>>(dinv, N);
    degree_kernel<<<blocks(E, B), B, 0, stream>>>(dst, dinv, E);
    dinv_kernel<<<blocks(N, B), B, 0, stream>>>(dinv, N);

    // ---- layer 1: relu(gcn(x, W1, b1)) -> l1 ----
    gemm_wmma_f32<128, 64><<<dim3(N / 16, 4), 32, 0, stream>>>(x, W1, hbuf);
    zero_f32<<<blocks((long long)N * 64, B), B, 0, stream>>>(agg, N * 64);
    scatter64<<<blocks((long long)E * 32, B), B, 0, stream>>>(hbuf, src, dst, dinv, agg, E);
    finalize_kernel<64, true><<<blocks((long long)N * 64, B), B, 0, stream>>>(
        agg, hbuf, dinv, b1, l1, N);

    // ---- layer 2: relu(gcn(l1, Wh, bh)) -> xlat (x_latent slice of d_out) ----
    gemm_wmma_f32<64, 64><<<dim3(N / 16, 4), 32, 0, stream>>>(l1, Wh, hbuf);
    zero_f32<<<blocks((long long)N * 64, B), B, 0, stream>>>(agg, N * 64);
    scatter64<<<blocks((long long)E * 32, B), B, 0, stream>>>(hbuf, src, dst, dinv, agg, E);
    finalize_kernel<64, true><<<blocks((long long)N * 64, B), B, 0, stream>>>(
        agg, hbuf, dinv, bh, xlat, N);

    // ---- layer 3: gcn(xlat, W2, b2) -> out (no relu) ----
    gemm_wmma_f32<64, 16><<<dim3(N / 16, 1), 32, 0, stream>>>(xlat, W2, hbuf);
    zero_f32<<<blocks((long long)N * 16, B), B, 0, stream>>>(agg, N * 16);
    scatter16<<<blocks((long long)E * 16, B), B, 0, stream>>>(hbuf, src, dst, dinv, agg, E);
    finalize_kernel<16, false><<<blocks((long long)N * 16, B), B, 0, stream>>>(
        agg, hbuf, dinv, b2, out, N);
}